// DyadBlock_52080773431509
// MI455X (gfx1250) — compile-verified
//
#include <hip/hip_runtime.h>

// ---------------------------------------------------------------------------
// DyadBlock: out[i] = W_lo[i] @ incl_i + W_up[i] @ (total - incl_i) + bias
//          = (W_lo[i]-W_up[i]) @ incl_i + W_up[i] @ total + bias
// DY=256 dyads, DO=DI=32, B=8192, fp32 throughout.
// Kernel 1: total[di][b] = sum_j x[j*DI+di][b]  (268 MB read, 1 MB write)
// Kernel 2: per 16-column tile, sweep dyads keeping incl in registers in
//           WMMA B-fragment layout; 32x v_wmma_f32_16x16x4_f32 per dyad.
// ---------------------------------------------------------------------------

#define DY 256
#define DO 32
#define DI 32
#define BN 8192

typedef __attribute__((ext_vector_type(2))) float v2f;
typedef __attribute__((ext_vector_type(8))) float v8f;

// ---------------- Kernel 1: column totals --------------------------------
__global__ __launch_bounds__(256) void dyad_total_kernel(
    const float* __restrict__ x, float* __restrict__ total) {
  // t = di*BN + b ; x[(j*DI+di)*BN + b] = x[j*DI*BN + t]
  const int t = blockIdx.x * 256 + threadIdx.x;
  const float* p = x + t;
  float s = 0.0f;
#pragma unroll 8
  for (int j = 0; j < DY; ++j) {
    s += __builtin_nontemporal_load(p + (size_t)j * (DI * BN));
  }
  total[t] = s;
}

// ---------------- Kernel 2: main fused sweep -----------------------------
__global__ __launch_bounds__(256) void dyad_main_kernel(
    const float* __restrict__ x,
    const float* __restrict__ w_upper,
    const float* __restrict__ w_lower,
    const float* __restrict__ bias,
    const float* __restrict__ total,
    float* __restrict__ out) {
  const int lane = threadIdx.x & 31;
  const int wave = threadIdx.x >> 5;
  const int tile = blockIdx.x * 8 + wave;   // 64 blocks * 8 waves = 512 tiles
  const int col0 = tile * 16;
  const int half = lane >> 4;               // lane group: 0 or 1
  const int nl   = lane & 15;               // N (and M) index within 16
  const int b    = col0 + nl;

  // B-fragment layout (32-bit 4x16): element (k,n) -> lane = n + 16*(k>>1),
  // vgpr = k&1. For k-step kk: k = kk*4 + half*2 + v.
  v2f tot[8];
  v2f incl[8];
#pragma unroll
  for (int kk = 0; kk < 8; ++kk) {
    const int di = kk * 4 + half * 2;
    tot[kk].x = total[(size_t)di * BN + b];
    tot[kk].y = total[(size_t)(di + 1) * BN + b];
    incl[kk].x = 0.0f;
    incl[kk].y = 0.0f;
  }

  for (int j = 0; j < DY; ++j) {
    const float* xj = x + (size_t)j * (DI * BN);

    // Prefetch next dyad's x rows for this lane (streaming hint).
    if (j + 1 < DY) {
      __builtin_prefetch(xj + (size_t)(DI * BN) + (size_t)(half * 2) * BN + b, 0, 1);
    }

    // Accumulate running prefix sum directly in B-fragment layout.
#pragma unroll
    for (int kk = 0; kk < 8; ++kk) {
      const float* xr = xj + (size_t)(kk * 4 + half * 2) * BN + b;
      incl[kk].x += __builtin_nontemporal_load(xr);
      incl[kk].y += __builtin_nontemporal_load(xr + BN);
    }

    // Weights for this dyad. A-fragment (32-bit 16x4): element (m,k) ->
    // lane = m + 16*(k>>1), vgpr = k&1. Lane loads contiguous float2.
    const float* WU = w_upper + (size_t)j * (DO * DI);
    const float* WL = w_lower + (size_t)j * (DO * DI);

    v8f acc0 = {};  // output rows o = 0..15   (M-tile 0)
    v8f acc1 = {};  // output rows o = 16..31  (M-tile 1)

#pragma unroll
    for (int kk = 0; kk < 8; ++kk) {
      const int krow = kk * 4 + half * 2;
      v2f wu0 = *(const v2f*)(WU + (size_t)nl * DI + krow);
      v2f wl0 = *(const v2f*)(WL + (size_t)nl * DI + krow);
      v2f wu1 = *(const v2f*)(WU + (size_t)(16 + nl) * DI + krow);
      v2f wl1 = *(const v2f*)(WL + (size_t)(16 + nl) * DI + krow);
      v2f wd0 = wl0 - wu0;
      v2f wd1 = wl1 - wu1;

      // acc += wd @ incl ; acc += wu @ total   (fp32 WMMA, K=4 per step)
      acc0 = __builtin_amdgcn_wmma_f32_16x16x4_f32(
          false, wd0, false, incl[kk], (short)0, acc0, false, false);
      acc0 = __builtin_amdgcn_wmma_f32_16x16x4_f32(
          false, wu0, false, tot[kk], (short)0, acc0, false, false);
      acc1 = __builtin_amdgcn_wmma_f32_16x16x4_f32(
          false, wd1, false, incl[kk], (short)0, acc1, false, false);
      acc1 = __builtin_amdgcn_wmma_f32_16x16x4_f32(
          false, wu1, false, tot[kk], (short)0, acc1, false, false);
    }

    // C-fragment layout: vgpr r, lane -> M = r + 8*half, N = nl.
    float* oj = out + (size_t)j * (DO * BN);
#pragma unroll
    for (int r = 0; r < 8; ++r) {
      const int o0 = r + half * 8;
      const int o1 = 16 + o0;
      __builtin_nontemporal_store(acc0[r] + bias[j * DO + o0],
                                  oj + (size_t)o0 * BN + b);
      __builtin_nontemporal_store(acc1[r] + bias[j * DO + o1],
                                  oj + (size_t)o1 * BN + b);
    }
  }
}

// ---------------- Host launch --------------------------------------------
extern "C" void kernel_launch(void* const* d_in, const int* in_sizes, int n_in,
                              void* d_out, int out_size, void* d_ws, size_t ws_size,
                              hipStream_t stream) {
  const float* x    = (const float*)d_in[0];
  const float* w_up = (const float*)d_in[1];
  const float* w_lo = (const float*)d_in[2];
  const float* bias = (const float*)d_in[3];
  float* out   = (float*)d_out;
  float* total = (float*)d_ws;  // DI*BN floats = 1 MB scratch

  dyad_total_kernel<<<(DI * BN) / 256, 256, 0, stream>>>(x, total);
  dyad_main_kernel<<<(BN / 16) / 8, 256, 0, stream>>>(x, w_up, w_lo, bias,
                                                      total, out);
}